// Critic_22617297781174
// MI455X (gfx1250) — compile-verified
//
#include <hip/hip_runtime.h>
#include <hip/hip_bf16.h>

#define F        20
#define TWO_F    40
#define R        20
#define N_LINKS  262144
#define N_PAIRS  4194304
#define N_TILES  (N_PAIRS / 16)

typedef __attribute__((ext_vector_type(2))) float v2f;
typedef __attribute__((ext_vector_type(8))) float v8f;

// D = A(16x4 f32) * B(4x16 f32) + C(16x16 f32)
static __device__ inline v8f wmma4(v2f a, v2f b, v8f c) {
    // 8 args: (neg_a, A, neg_b, B, c_mod, C, reuse_a, reuse_b)
    return __builtin_amdgcn_wmma_f32_16x16x4_f32(
        false, a, false, b, (short)0, c, false, false);
}

// Branchless SELU: exact on both sides, no exec-mask divergence.
//   x > 0 : scale * (x + alpha*(exp(0)-1))   = scale * x
//   x <= 0: scale * (0 + alpha*(exp(x)-1))
static __device__ inline float selu_f(float x) {
    const float kScale = 1.0507009873554805f;
    const float kAlpha = 1.6732632423543772f;
    float e = __expf(fminf(x, 0.f)) - 1.f;
    return kScale * fmaf(kAlpha, e, fmaxf(x, 0.f));
}

// ---------------------------------------------------------------------------
// Message MLP + scatter-add, WMMA f32 16x16x4.
// Each wave: 16 pairs per tile. A row = concat(state[first[p]], state[second[p]]) (K=40).
// A-frag layout (ISA 7.12.2, 32-bit A 16x4): lane l -> row M=l%16, K offset 2*(l>>4).
// B[k][n] = W_msg[n][k]; two N-tiles (n 0..15, 16..31; valid n<20).
// C/D layout: VGPR r, lane l -> M = r + 8*(l>>4), N = l%16 (+16 for tile 1).
// Bias is constant per lane (N = lane%16) -> folded into C init.
// ---------------------------------------------------------------------------
__global__ void __launch_bounds__(256)
msg_kernel(const float* __restrict__ state,
           const int*   __restrict__ first,
           const int*   __restrict__ second,
           const float* __restrict__ W_msg,   // [F][2F] row-major
           const float* __restrict__ b_msg,   // [F]
           float*       __restrict__ agg)     // [N_LINKS][F]
{
    const int lane  = threadIdx.x & 31;
    const int half  = lane >> 4;     // 0 or 1
    const int mrow  = lane & 15;     // row within 16-pair tile / N within tile
    const int waveId  = (blockIdx.x * blockDim.x + threadIdx.x) >> 5;
    const int nWaves  = (gridDim.x * blockDim.x) >> 5;

    // Preload B fragments: 10 K-steps x 2 N-tiles.
    // frag v0 = B[k0 + 2*half][n], v1 = B[k0 + 1 + 2*half][n]
    v2f bfrag[2][10];
    #pragma unroll
    for (int t = 0; t < 2; ++t) {
        const int n = 16 * t + mrow;
        const bool nv = (n < F);
        #pragma unroll
        for (int ks = 0; ks < 10; ++ks) {
            const int k = 4 * ks + 2 * half;
            v2f b;
            b.x = nv ? W_msg[n * TWO_F + k]     : 0.f;
            b.y = nv ? W_msg[n * TWO_F + k + 1] : 0.f;
            bfrag[t][ks] = b;
        }
    }
    const float bias0 = b_msg[mrow];                        // n = mrow < 20 always
    const float bias1 = (16 + mrow < F) ? b_msg[16 + mrow] : 0.f;

    for (int tile = waveId; tile < N_TILES; tile += nWaves) {
        const int p  = tile * 16 + mrow;
        const int fi = first[p];
        const int se = second[p];
        const float* rowF = state + (size_t)fi * F;
        const float* rowS = state + (size_t)se * F;

        // Fold bias into accumulator init (N is constant per lane).
        v8f c0 = {bias0, bias0, bias0, bias0, bias0, bias0, bias0, bias0};
        v8f c1 = {bias1, bias1, bias1, bias1, bias1, bias1, bias1, bias1};

        #pragma unroll
        for (int ks = 0; ks < 10; ++ks) {
            const int k = 4 * ks + 2 * half;   // even; (k,k+1) never straddles F=20
            const float* rp = (k < F) ? (rowF + k) : (rowS + (k - F));
            v2f a;
            a.x = rp[0];
            a.y = rp[1];
            c0 = wmma4(a, bfrag[0][ks], c0);
            c1 = wmma4(a, bfrag[1][ks], c1);
        }

        // SELU + scatter-add into agg[second[pair]]
        #pragma unroll
        for (int r = 0; r < 8; ++r) {
            const int m   = r + 8 * half;            // pair row of this D element
            const int idx = __shfl(se, m, 32);       // second[] of pair tile*16+m
            float v0 = selu_f(c0[r]);
            unsafeAtomicAdd(&agg[(size_t)idx * F + mrow], v0);
            if (mrow < 4) {
                float v1 = selu_f(c1[r]);
                unsafeAtomicAdd(&agg[(size_t)idx * F + 16 + mrow], v1);
            }
        }
    }
}

// ---------------------------------------------------------------------------
// GRU cell, one thread per link. Weight addresses are wave-uniform -> scalar path.
// ---------------------------------------------------------------------------
__global__ void __launch_bounds__(256)
gru_kernel(const float* __restrict__ agg,
           const float* hin,           // may alias hout (in-place after iter 0)
           const float* __restrict__ W_ih,  // [3F][F]
           const float* __restrict__ W_hh,  // [3F][F]
           const float* __restrict__ b_ih,  // [3F]
           const float* __restrict__ b_hh,  // [3F]
           float* hout)
{
    const int link = blockIdx.x * blockDim.x + threadIdx.x;
    if (link >= N_LINKS) return;

    float m[F], h[F];
    const float* ar = agg + (size_t)link * F;
    const float* hr = hin + (size_t)link * F;
    #pragma unroll
    for (int k = 0; k < F; ++k) { m[k] = ar[k]; h[k] = hr[k]; }

    float* orow = hout + (size_t)link * F;
    for (int j = 0; j < F; ++j) {
        float ir = b_ih[j], iz = b_ih[F + j], inn = b_ih[2 * F + j];
        float hrv = b_hh[j], hz = b_hh[F + j], hn = b_hh[2 * F + j];
        #pragma unroll
        for (int k = 0; k < F; ++k) {
            ir  += W_ih[j * F + k]           * m[k];
            iz  += W_ih[(F + j) * F + k]     * m[k];
            inn += W_ih[(2 * F + j) * F + k] * m[k];
            hrv += W_hh[j * F + k]           * h[k];
            hz  += W_hh[(F + j) * F + k]     * h[k];
            hn  += W_hh[(2 * F + j) * F + k] * h[k];
        }
        const float r = 1.f / (1.f + __expf(-(ir + hrv)));
        const float z = 1.f / (1.f + __expf(-(iz + hz)));
        const float n = tanhf(inn + r * hn);
        orow[j] = (1.f - z) * n + z * h[j];
    }
}

// ---------------------------------------------------------------------------
// feature[j] = sum over links of state[link][j]
// ---------------------------------------------------------------------------
__global__ void __launch_bounds__(256)
reduce_kernel(const float* __restrict__ state, float* __restrict__ feat)
{
    __shared__ float sf[F];
    if (threadIdx.x < F) sf[threadIdx.x] = 0.f;
    __syncthreads();

    float loc[F];
    #pragma unroll
    for (int j = 0; j < F; ++j) loc[j] = 0.f;

    for (int link = blockIdx.x * blockDim.x + threadIdx.x; link < N_LINKS;
         link += gridDim.x * blockDim.x) {
        const float* r = state + (size_t)link * F;
        #pragma unroll
        for (int j = 0; j < F; ++j) loc[j] += r[j];
    }
    #pragma unroll
    for (int j = 0; j < F; ++j) atomicAdd(&sf[j], loc[j]);
    __syncthreads();
    if (threadIdx.x < F) unsafeAtomicAdd(&feat[threadIdx.x], sf[threadIdx.x]);
}

// ---------------------------------------------------------------------------
// Readout MLP, single block of 32 threads.
// ---------------------------------------------------------------------------
__global__ void readout_kernel(const float* __restrict__ feat,
                               const float* __restrict__ W_r1, const float* __restrict__ b_r1,
                               const float* __restrict__ W_r2, const float* __restrict__ b_r2,
                               const float* __restrict__ W_out, const float* __restrict__ b_out,
                               float* __restrict__ out)
{
    __shared__ float f[F], h1[R], h2[R];
    const int t = threadIdx.x;
    if (t < F) f[t] = feat[t];
    __syncthreads();
    if (t < R) {
        float a = b_r1[t];
        for (int k = 0; k < F; ++k) a += W_r1[t * F + k] * f[k];
        h1[t] = selu_f(a);
    }
    __syncthreads();
    if (t < R) {
        float a = b_r2[t];
        for (int k = 0; k < R; ++k) a += W_r2[t * R + k] * h1[k];
        h2[t] = selu_f(a);
    }
    __syncthreads();
    if (t == 0) {
        float a = b_out[0];
        for (int k = 0; k < R; ++k) a += W_out[k] * h2[k];
        out[0] = a;
    }
}

extern "C" void kernel_launch(void* const* d_in, const int* in_sizes, int n_in,
                              void* d_out, int out_size, void* d_ws, size_t ws_size,
                              hipStream_t stream) {
    (void)in_sizes; (void)n_in; (void)out_size; (void)ws_size;
    const float* link_state = (const float*)d_in[0];
    const int*   first      = (const int*)  d_in[1];
    const int*   second     = (const int*)  d_in[2];
    // d_in[3] = state_dim (scalar), unused
    const float* W_msg = (const float*)d_in[4];
    const float* b_msg = (const float*)d_in[5];
    const float* W_ih  = (const float*)d_in[6];
    const float* W_hh  = (const float*)d_in[7];
    const float* b_ih  = (const float*)d_in[8];
    const float* b_hh  = (const float*)d_in[9];
    const float* W_r1  = (const float*)d_in[10];
    const float* b_r1  = (const float*)d_in[11];
    const float* W_r2  = (const float*)d_in[12];
    const float* b_r2  = (const float*)d_in[13];
    const float* W_out = (const float*)d_in[14];
    const float* b_out = (const float*)d_in[15];
    float* out = (float*)d_out;

    float* state = (float*)d_ws;                       // [N_LINKS*F]
    float* agg   = state + (size_t)N_LINKS * F;        // [N_LINKS*F]
    float* feat  = agg   + (size_t)N_LINKS * F;        // [F]

    const int T = 4;
    for (int it = 0; it < T; ++it) {
        const float* src = (it == 0) ? link_state : state;
        hipMemsetAsync(agg, 0, (size_t)N_LINKS * F * sizeof(float), stream);
        msg_kernel<<<2048, 256, 0, stream>>>(src, first, second, W_msg, b_msg, agg);
        gru_kernel<<<N_LINKS / 256, 256, 0, stream>>>(agg, src, W_ih, W_hh, b_ih, b_hh, state);
    }
    hipMemsetAsync(feat, 0, F * sizeof(float), stream);
    reduce_kernel<<<256, 256, 0, stream>>>(state, feat);
    readout_kernel<<<1, 32, 0, stream>>>(feat, W_r1, b_r1, W_r2, b_r2, W_out, b_out, out);
}